// KipfAndWillingConv_24464133718385
// MI455X (gfx1250) — compile-verified
//
#include <hip/hip_runtime.h>

typedef float v2f __attribute__((ext_vector_type(2)));
typedef float v8f __attribute__((ext_vector_type(8)));

#define N_FEAT 128          // features == filters == 128

// ---------------------------------------------------------------------------
// Kernel 1: zero the output (harness poisons d_out with 0xAA before timing)
// ---------------------------------------------------------------------------
__global__ __launch_bounds__(256)
void gcn_zero(float4* __restrict__ out4, int n4) {
    int i = blockIdx.x * 256 + threadIdx.x;
    if (i < n4) out4[i] = make_float4(0.f, 0.f, 0.f, 0.f);
}

// ---------------------------------------------------------------------------
// Kernel 2: XF = x @ filters using V_WMMA_F32_16X16X4_F32 (exact fp32 math).
// grid.x = M/16 tiles; block = 256 threads = 8 waves; wave w owns N-tile w.
// A tile (16x128, 8KB) staged in LDS (stride 130 avoids bank conflicts);
// filters (64KB) read via L2 (permanently resident there).
// ---------------------------------------------------------------------------
__global__ __launch_bounds__(256)
void gcn_gemm_wmma(const float* __restrict__ x,
                   const float* __restrict__ filt,
                   float* __restrict__ xf) {
    __shared__ float As[16 * 130];

    const int tid  = threadIdx.x;
    const long row0 = (long)blockIdx.x * 16;

    // Cooperative coalesced load of the 16x128 A tile into LDS.
    #pragma unroll
    for (int i = 0; i < 8; ++i) {
        int f = i * 256 + tid;
        int r = f >> 7, c = f & 127;
        As[r * 130 + c] = x[(row0 + r) * N_FEAT + c];
    }
    __syncthreads();

    const int wave = tid >> 5;
    const int lane = tid & 31;
    const int m    = lane & 15;   // A row / B col / D col
    const int kg   = lane >> 4;   // which K-pair this lane half holds
    const int n0   = wave * 16;   // N-tile origin for this wave

    v8f c = {};
    for (int k = 0; k < N_FEAT; k += 4) {
        const int kk = k + 2 * kg;
        v2f a, b;
        // A 16x4 layout: lanes 0-15 hold K=0,1 ; lanes 16-31 hold K=2,3
        a.x = As[m * 130 + kk];
        a.y = As[m * 130 + kk + 1];
        // B 4x16 layout (N striped across lanes, K split across lane halves)
        b.x = filt[(long)kk * N_FEAT + n0 + m];
        b.y = filt[(long)(kk + 1) * N_FEAT + n0 + m];
        // 8 args: (neg_a, A, neg_b, B, c_mod, C, reuse_a, reuse_b)
        c = __builtin_amdgcn_wmma_f32_16x16x4_f32(
                false, a, false, b, (short)0, c, false, false);
    }

    // D 16x16 f32: VGPR r holds (M = r + 8*kg, N = m)
    #pragma unroll
    for (int r = 0; r < 8; ++r) {
        xf[(row0 + r + 8 * kg) * N_FEAT + n0 + m] = c[r];
    }
}

// ---------------------------------------------------------------------------
// Kernel 3: out[d] += w_e * XF[s] over edges, exploiting sorted edge_dst.
// One wave owns 128 contiguous edges; lane owns features [4*lane, 4*lane+4).
// Register accumulation per destination segment; atomicAdd only at segment
// boundaries (avg degree 32 -> ~5 flushes/wave). Edge reads are wave-uniform
// so the compiler scalarizes them (s_load); XF gathers are coalesced b128
// loads that hit L2 (XF = 51.2MB << 192MB L2).
// ---------------------------------------------------------------------------
__global__ __launch_bounds__(256)
void gcn_scatter(const float* __restrict__ xf,
                 const int* __restrict__ esrc,
                 const int* __restrict__ edst,
                 const float* __restrict__ ew,
                 float* __restrict__ out,
                 int n_edges) {
    const int EPW  = 128;  // edges per wave
    const int wid  = blockIdx.x * (blockDim.x >> 5) + (threadIdx.x >> 5);
    const int lane = threadIdx.x & 31;

    long e0 = (long)wid * EPW;
    if (e0 >= n_edges) return;
    long e1 = e0 + EPW;
    if (e1 > n_edges) e1 = n_edges;

    const float4* __restrict__ xf4 = (const float4*)xf;

    int   cur_d = edst[e0];
    float4 acc  = make_float4(0.f, 0.f, 0.f, 0.f);

    for (long e = e0; e < e1; ++e) {
        int   s = esrc[e];
        int   d = edst[e];
        float w = ew[e];
        if (d != cur_d) {                      // wave-uniform branch
            float* p = &out[(long)cur_d * N_FEAT + lane * 4];
            atomicAdd(p + 0, acc.x);
            atomicAdd(p + 1, acc.y);
            atomicAdd(p + 2, acc.z);
            atomicAdd(p + 3, acc.w);
            acc   = make_float4(0.f, 0.f, 0.f, 0.f);
            cur_d = d;
        }
        float4 v = xf4[(long)s * (N_FEAT / 4) + lane];   // global_load_b128
        acc.x += w * v.x;
        acc.y += w * v.y;
        acc.z += w * v.z;
        acc.w += w * v.w;
    }
    float* p = &out[(long)cur_d * N_FEAT + lane * 4];
    atomicAdd(p + 0, acc.x);
    atomicAdd(p + 1, acc.y);
    atomicAdd(p + 2, acc.z);
    atomicAdd(p + 3, acc.w);
}

// ---------------------------------------------------------------------------
extern "C" void kernel_launch(void* const* d_in, const int* in_sizes, int n_in,
                              void* d_out, int out_size, void* d_ws, size_t ws_size,
                              hipStream_t stream) {
    const float* x    = (const float*)d_in[0];   // [N_NODES, 128]
    const float* filt = (const float*)d_in[1];   // [128, 128]
    const int*   esrc = (const int*)d_in[2];     // [E]
    const int*   edst = (const int*)d_in[3];     // [E] (sorted)
    const float* ew   = (const float*)d_in[4];   // [E]
    float* out = (float*)d_out;
    float* xf  = (float*)d_ws;                   // [N_NODES, 128] scratch

    const int n_nodes = in_sizes[0] / N_FEAT;    // 100000
    const int n_edges = in_sizes[2];             // 3200000

    // 1) out = 0
    const int n4 = n_nodes * N_FEAT / 4;
    gcn_zero<<<(n4 + 255) / 256, 256, 0, stream>>>((float4*)out, n4);

    // 2) XF = x @ filters  (fp32 WMMA)
    gcn_gemm_wmma<<<n_nodes / 16, 256, 0, stream>>>(x, filt, xf);

    // 3) segment scatter-add
    const int waves  = (n_edges + 127) / 128;
    const int blocks = (waves + 7) / 8;          // 8 waves / 256-thread block
    gcn_scatter<<<blocks, 256, 0, stream>>>(xf, esrc, edst, ew, out, n_edges);
}